// GenericConv_3934190044274
// MI455X (gfx1250) — compile-verified
//
#include <hip/hip_runtime.h>

#define N_NODES  100000
#define N_EDGES  1600000
#define F_DIM    128
#define N_GRAPHS 64

typedef float v2f __attribute__((ext_vector_type(2)));
typedef float v8f __attribute__((ext_vector_type(8)));

// ---------------- degree / normalization ----------------
__global__ __launch_bounds__(256) void init_deg_kernel(float* deg) {
    int i = blockIdx.x * 256 + threadIdx.x;
    if (i < N_NODES) deg[i] = 1.0f;               // self-loop fill_value = 1.0
}

__global__ __launch_bounds__(256) void deg_scatter_kernel(const int* __restrict__ ei,
                                                          const float* __restrict__ ew,
                                                          float* deg) {
    int e = blockIdx.x * 256 + threadIdx.x;
    if (e < N_EDGES) atomicAdd(&deg[ei[N_EDGES + e]], ew[e]);   // dst row
}

__global__ __launch_bounds__(256) void rsqrt_kernel(float* deg_to_dinv) {
    int i = blockIdx.x * 256 + threadIdx.x;
    if (i < N_NODES) deg_to_dinv[i] = rsqrtf(deg_to_dinv[i]);
}

// ---------------- fp32 WMMA GEMM: out[N,128] = relu?(in[N,128]) @ W[128,128] ----------------
// One wave32 per 16-row M-tile; 8 accumulator tiles cover all 128 output columns.
// A fragment (16x4 f32): lane L holds row M = L%16, K = k + 2*(L>>4) + {0,1} in v[0],v[1].
// B fragment (4x16 f32): lane L holds col N = L%16, K = k + 2*(L>>4) + {0,1} in v[0],v[1].
// C/D (16x16 f32): lane L col N = L%16, VGPR r holds row M = r + 8*(L>>4).
template <bool RELU_IN>
__global__ __launch_bounds__(32) void gemm_wmma_kernel(const float* __restrict__ A,
                                                       const float* __restrict__ W,
                                                       float* __restrict__ out) {
    const int lane = threadIdx.x;
    const int lo = lane & 15;
    const int hi = lane >> 4;
    const int m0 = blockIdx.x * 16;

    v8f acc[8];
    const v8f vzero = {0.f, 0.f, 0.f, 0.f, 0.f, 0.f, 0.f, 0.f};
#pragma unroll
    for (int n = 0; n < 8; ++n) acc[n] = vzero;

    const float* aptr = A + (size_t)(m0 + lo) * F_DIM;

    for (int k = 0; k < F_DIM; k += 4) {
        const int ka = k + 2 * hi;
        v2f a;
        a.x = aptr[ka];
        a.y = aptr[ka + 1];
        if (RELU_IN) { a.x = fmaxf(a.x, 0.f); a.y = fmaxf(a.y, 0.f); }
        const float* wp0 = W + (size_t)ka * F_DIM + lo;
        const float* wp1 = wp0 + F_DIM;
#pragma unroll
        for (int n = 0; n < 8; ++n) {
            v2f b;
            b.x = wp0[n * 16];
            b.y = wp1[n * 16];
            acc[n] = __builtin_amdgcn_wmma_f32_16x16x4_f32(
                false, a, false, b, (short)0, acc[n], false, false);
        }
    }

    float* op = out + (size_t)m0 * F_DIM;
#pragma unroll
    for (int n = 0; n < 8; ++n)
#pragma unroll
        for (int r = 0; r < 8; ++r)
            op[(size_t)(r + 8 * hi) * F_DIM + n * 16 + lo] = acc[n][r];
}

// ---------------- agg init: agg = dinv^2 * h + bias ----------------
__global__ __launch_bounds__(256) void init_agg_kernel(const float* __restrict__ h,
                                                       const float* __restrict__ dinv,
                                                       const float* __restrict__ bias,
                                                       float* __restrict__ agg) {
    size_t idx = (size_t)blockIdx.x * 256 + threadIdx.x;
    if (idx >= (size_t)N_NODES * F_DIM) return;
    int i = (int)(idx >> 7);
    int f = (int)(idx & 127);
    float d = dinv[i];
    agg[idx] = d * d * h[idx] + bias[f];
}

// ---------------- edge scatter: agg[dst] += norm(e) * h[src]  (1 wave per edge) ----------------
__global__ __launch_bounds__(256) void edge_scatter_kernel(const int* __restrict__ ei,
                                                           const float* __restrict__ ew,
                                                           const float* __restrict__ dinv,
                                                           const float* __restrict__ h,
                                                           float* __restrict__ agg) {
    const int lane = threadIdx.x & 31;
    const int e = blockIdx.x * 8 + (threadIdx.x >> 5);
    if (e >= N_EDGES) return;
    const int src = ei[e];
    const int dst = ei[N_EDGES + e];
    const float nrm = dinv[src] * ew[e] * dinv[dst];
    const float4 hv = *(const float4*)(h + (size_t)src * F_DIM + lane * 4);
    float* ap = agg + (size_t)dst * F_DIM + lane * 4;
    atomicAdd(ap + 0, nrm * hv.x);
    atomicAdd(ap + 1, nrm * hv.y);
    atomicAdd(ap + 2, nrm * hv.z);
    atomicAdd(ap + 3, nrm * hv.w);
}

// ---------------- pooling ----------------
__global__ __launch_bounds__(256) void zero_out_kernel(float* out) {
    int i = blockIdx.x * 256 + threadIdx.x;
    if (i < N_GRAPHS * F_DIM) out[i] = 0.0f;
}

// segment_sum of relu(h) by batch id (1 wave per node)
__global__ __launch_bounds__(256) void pool_kernel(const float* __restrict__ h,
                                                   const int* __restrict__ batch,
                                                   float* __restrict__ out) {
    const int lane = threadIdx.x & 31;
    const int i = blockIdx.x * 8 + (threadIdx.x >> 5);
    if (i >= N_NODES) return;
    const int g = batch[i];
    const float4 hv = *(const float4*)(h + (size_t)i * F_DIM + lane * 4);
    float* op = out + (size_t)g * F_DIM + lane * 4;
    atomicAdd(op + 0, fmaxf(hv.x, 0.f));
    atomicAdd(op + 1, fmaxf(hv.y, 0.f));
    atomicAdd(op + 2, fmaxf(hv.z, 0.f));
    atomicAdd(op + 3, fmaxf(hv.w, 0.f));
}

extern "C" void kernel_launch(void* const* d_in, const int* in_sizes, int n_in,
                              void* d_out, int out_size, void* d_ws, size_t ws_size,
                              hipStream_t stream) {
    (void)in_sizes; (void)n_in; (void)out_size; (void)ws_size;
    const float* x     = (const float*)d_in[0];
    const int*   ei    = (const int*)d_in[1];
    const float* ew    = (const float*)d_in[2];
    const int*   batch = (const int*)d_in[3];
    const float* W1    = (const float*)d_in[4];
    const float* b1    = (const float*)d_in[5];
    const float* W2    = (const float*)d_in[6];
    const float* b2    = (const float*)d_in[7];
    float* out = (float*)d_out;

    // workspace layout (16B aligned offsets)
    char* ws = (char*)d_ws;
    const size_t dinv_bytes = ((size_t)N_NODES * 4 + 255) & ~(size_t)255;   // 400128
    const size_t mat_bytes  = (size_t)N_NODES * F_DIM * 4;                  // 51.2 MB
    float* dinv = (float*)ws;
    float* bufA = (float*)(ws + dinv_bytes);                 // h = in @ W
    float* bufB = (float*)(ws + dinv_bytes + mat_bytes);     // agg / layer output

    const int nblk_nodes = (N_NODES + 255) / 256;
    const int nblk_edges = (N_EDGES + 255) / 256;
    const int nblk_nf    = (int)(((size_t)N_NODES * F_DIM + 255) / 256);
    const int nblk_escat = (N_EDGES + 7) / 8;
    const int nblk_pool  = (N_NODES + 7) / 8;
    const int nblk_gemm  = N_NODES / 16;   // 6250, exact

    // normalization
    init_deg_kernel<<<nblk_nodes, 256, 0, stream>>>(dinv);
    deg_scatter_kernel<<<nblk_edges, 256, 0, stream>>>(ei, ew, dinv);
    rsqrt_kernel<<<nblk_nodes, 256, 0, stream>>>(dinv);

    // layer 1
    gemm_wmma_kernel<false><<<nblk_gemm, 32, 0, stream>>>(x, W1, bufA);
    init_agg_kernel<<<nblk_nf, 256, 0, stream>>>(bufA, dinv, b1, bufB);
    edge_scatter_kernel<<<nblk_escat, 256, 0, stream>>>(ei, ew, dinv, bufA, bufB);

    // layer 2 (ReLU of layer-1 output fused into GEMM A-loads)
    gemm_wmma_kernel<true><<<nblk_gemm, 32, 0, stream>>>(bufB, W2, bufA);
    init_agg_kernel<<<nblk_nf, 256, 0, stream>>>(bufA, dinv, b2, bufB);
    edge_scatter_kernel<<<nblk_escat, 256, 0, stream>>>(ei, ew, dinv, bufA, bufB);

    // pool (ReLU fused)
    zero_out_kernel<<<(N_GRAPHS * F_DIM + 255) / 256, 256, 0, stream>>>(out);
    pool_kernel<<<nblk_pool, 256, 0, stream>>>(bufB, batch, out);
}